// EdgeConvBlock_34892314313403
// MI455X (gfx1250) — compile-verified
//
#include <hip/hip_runtime.h>
#include <math.h>

#define BB 8
#define NN 4096
#define CC 64
#define DD 64
#define KK 16
#define JCHUNK 2048   // distance columns per LDS chunk: 16*2048*4 = 128 KB (< 320 KB WGP LDS)

typedef float v2f __attribute__((ext_vector_type(2)));
typedef float v8f __attribute__((ext_vector_type(8)));

__device__ __forceinline__ v8f zero8() {
    v8f z = {0.f, 0.f, 0.f, 0.f, 0.f, 0.f, 0.f, 0.f};
    return z;
}

__device__ __forceinline__ float gelu_exact(float x) {
    // 0.5*x*(1+erf(x/sqrt(2))) — matches jax.nn.gelu(approximate=False)
    return 0.5f * x * (1.0f + erff(x * 0.70710678118654752f));
}

// ---------------------------------------------------------------------------
// Kernel A: per-row squared norms  sq[b*N+i] = sum_c f^2
// ---------------------------------------------------------------------------
__global__ __launch_bounds__(256) void sqnorm_kernel(const float* __restrict__ F,
                                                     float* __restrict__ sq) {
    int i = blockIdx.x * 256 + threadIdx.x;           // 0..B*N-1
    const float* row = F + (size_t)i * CC;
    float s = 0.f;
#pragma unroll
    for (int c = 0; c < CC; c += 4) {
        float4 v = *(const float4*)(row + c);
        s += v.x * v.x + v.y * v.y + v.z * v.z + v.w * v.w;
    }
    sq[i] = s;
}

// ---------------------------------------------------------------------------
// Kernel B: fused distance (f32 WMMA Gram) + top-16 selection.
// One workgroup (16 waves) per (batch, 16-row i-tile). Distances for a
// 16 x JCHUNK block live in LDS; selection is register-resident per lane.
// ---------------------------------------------------------------------------
__global__ __launch_bounds__(512) void knn_kernel(const float* __restrict__ F,
                                                  const float* __restrict__ sq,
                                                  int* __restrict__ idxbuf) {
    __shared__ float s_d2[16 * JCHUNK];

    const int b    = blockIdx.x >> 8;                 // / 256 i-tiles
    const int i0   = (blockIdx.x & 255) << 4;
    const int tid  = threadIdx.x;
    const int lane = tid & 31;
    const int wv   = tid >> 5;                        // 16 waves
    const int half = lane >> 4;
    const int lm   = lane & 15;

    const float* Fb  = F  + (size_t)b * NN * CC;
    const float* sqb = sq + (size_t)b * NN;

    // A fragments: F_i tile, 16 rows x 64 K, 16 ksteps of 16x4
    v2f afrag[16];
#pragma unroll
    for (int t = 0; t < 16; ++t)
        afrag[t] = *(const v2f*)(Fb + (size_t)(i0 + lm) * CC + 4 * t + 2 * half);

    // per-lane row norms for the 8 output rows this lane owns
    float sqi_r[8];
#pragma unroll
    for (int r = 0; r < 8; ++r) sqi_r[r] = sqb[i0 + r + 8 * half];

    // lane-local sorted top-16 (ascending), static-indexed -> registers only
    float bestv[16];
    int   besti[16];
#pragma unroll
    for (int u = 0; u < 16; ++u) { bestv[u] = 3.4e38f; besti[u] = 0; }

    for (int ch = 0; ch < NN / JCHUNK; ++ch) {
        // ---- stage 1: distances for columns [ch*JCHUNK, (ch+1)*JCHUNK) ----
        for (int u = 0; u < (JCHUNK / 16) / 16; ++u) {        // j-tiles per wave
            const int jt = (JCHUNK / 16) * ch + wv + 16 * u;
            const int j0 = jt * 16;
            v2f bfrag[16];
#pragma unroll
            for (int t = 0; t < 16; ++t)
                bfrag[t] = *(const v2f*)(Fb + (size_t)(j0 + lm) * CC + 4 * t + 2 * half);
            v8f acc = zero8();
#pragma unroll
            for (int t = 0; t < 16; ++t)
                acc = __builtin_amdgcn_wmma_f32_16x16x4_f32(
                    false, afrag[t], false, bfrag[t], (short)0, acc, false, false);

            const float sqj = sqb[j0 + lm];
            const int   jl0 = j0 - ch * JCHUNK;
#pragma unroll
            for (int r = 0; r < 8; ++r) {
                const int row = r + 8 * half;
                float v = sqi_r[r] + sqj - 2.0f * acc[r];
                if (j0 + lm == i0 + row) v = 3.0e38f;          // diagonal -> +inf
                s_d2[row * JCHUNK + jl0 + lm] = v;
            }
        }
        __syncthreads();

        // ---- stage 2: wave wv scans row wv, accumulates lane-local top-16 ----
        {
            const float* rp = s_d2 + wv * JCHUNK;
            for (int t = 0; t < JCHUNK / 32; ++t) {
                const int   jl  = lane + 32 * t;
                const float val = rp[jl];
                if (val < bestv[15]) {
                    float cv = val;
                    int   ci = ch * JCHUNK + jl;
#pragma unroll
                    for (int u = 0; u < 16; ++u) {             // sorted insertion ladder
                        if (cv < bestv[u]) {
                            float tv = bestv[u]; int ti = besti[u];
                            bestv[u] = cv; besti[u] = ci;
                            cv = tv; ci = ti;
                        }
                    }
                }
            }
        }
        __syncthreads();
    }

    // ---- merge 32 lane-local lists -> global top-16 for row (i0+wv) ----
    int* orow = idxbuf + (size_t)(b * NN + i0 + wv) * KK;
    for (int s = 0; s < KK; ++s) {
        float v  = bestv[0];
        int   id = besti[0];
        int   wl = lane;
#pragma unroll
        for (int off = 16; off > 0; off >>= 1) {               // wave32 argmin
            float ov  = __shfl_xor(v, off, 32);
            int   oi  = __shfl_xor(id, off, 32);
            int   owl = __shfl_xor(wl, off, 32);
            if (ov < v || (ov == v && owl < wl)) { v = ov; id = oi; wl = owl; }
        }
        if (lane == s) orow[s] = id;
        if (lane == wl) {                                      // winner pops its head
#pragma unroll
            for (int u = 0; u < 15; ++u) { bestv[u] = bestv[u + 1]; besti[u] = besti[u + 1]; }
            bestv[15] = 3.4e38f;
        }
    }
}

// ---------------------------------------------------------------------------
// Kernel C: P = F @ W1b ; Q = F @ (W1a - W1b) + b1   (per-point factorization)
// One block per 16-row tile; wave w handles 16-column tile w of both outputs.
// ---------------------------------------------------------------------------
__global__ __launch_bounds__(128) void pq_kernel(const float* __restrict__ F,
                                                 const float* __restrict__ W1,
                                                 const float* __restrict__ b1,
                                                 float* __restrict__ P,
                                                 float* __restrict__ Q) {
    const int b    = blockIdx.x >> 8;
    const int i0   = (blockIdx.x & 255) << 4;
    const int lane = threadIdx.x & 31;
    const int wv   = threadIdx.x >> 5;                 // 4 waves -> 4 col tiles
    const int half = lane >> 4;
    const int lm   = lane & 15;
    const int n0   = wv * 16;

    const float* Fb = F + (size_t)b * NN * CC;
    v2f afrag[16];
#pragma unroll
    for (int t = 0; t < 16; ++t)
        afrag[t] = *(const v2f*)(Fb + (size_t)(i0 + lm) * CC + 4 * t + 2 * half);

    v8f accP = zero8();
    v8f accQ = zero8();
#pragma unroll
    for (int t = 0; t < 16; ++t) {
        const int k0 = 4 * t + 2 * half;
        const float wb0 = W1[(size_t)(CC + k0) * DD + n0 + lm];       // W1b rows
        const float wb1 = W1[(size_t)(CC + k0 + 1) * DD + n0 + lm];
        const float wa0 = W1[(size_t)(k0) * DD + n0 + lm];            // W1a rows
        const float wa1 = W1[(size_t)(k0 + 1) * DD + n0 + lm];
        v2f bP, bQ;
        bP.x = wb0;        bP.y = wb1;
        bQ.x = wa0 - wb0;  bQ.y = wa1 - wb1;
        accP = __builtin_amdgcn_wmma_f32_16x16x4_f32(false, afrag[t], false, bP,
                                                     (short)0, accP, false, false);
        accQ = __builtin_amdgcn_wmma_f32_16x16x4_f32(false, afrag[t], false, bQ,
                                                     (short)0, accQ, false, false);
    }
    const float bias = b1[n0 + lm];
#pragma unroll
    for (int r = 0; r < 8; ++r) {
        const int row = r + 8 * half;
        const size_t o = (size_t)(b * NN + i0 + row) * DD + n0 + lm;
        P[o] = accP[r];
        Q[o] = accQ[r] + bias;
    }
}

// ---------------------------------------------------------------------------
// Kernel D: per point — h1[k,:] = gelu(Q[i] + P[idx[k]]) (16x64, in A-fragment
// layout), h2 = gelu(h1 @ W2 + b2), out = max over the 16 neighbors.
// One wave per point; max-pool via per-lane reg max + shfl_xor(16).
// ---------------------------------------------------------------------------
__global__ __launch_bounds__(256) void edge_mlp_kernel(const float* __restrict__ P,
                                                       const float* __restrict__ Q,
                                                       const int* __restrict__ idxbuf,
                                                       const float* __restrict__ W2,
                                                       const float* __restrict__ b2,
                                                       float* __restrict__ out) {
    const int lane = threadIdx.x & 31;
    const int wv   = threadIdx.x >> 5;                 // 8 waves per block
    const int half = lane >> 4;
    const int lm   = lane & 15;
    const int p    = blockIdx.x * 8 + wv;              // point id, 0..B*N-1
    const int b    = p >> 12;                          // / N

    const int nb = idxbuf[(size_t)p * KK + lm];        // lanes l and l+16 share neighbor lm
    const float* Qp = Q + (size_t)p * DD;
    const float* Pp = P + (size_t)(b * NN + nb) * DD;

    // A fragments of h1 (M = neighbor, K = feature dim)
    v2f afrag[16];
#pragma unroll
    for (int t = 0; t < 16; ++t) {
        const int d0 = 4 * t + 2 * half;
        v2f q  = *(const v2f*)(Qp + d0);
        v2f pv = *(const v2f*)(Pp + d0);
        afrag[t].x = gelu_exact(q.x + pv.x);
        afrag[t].y = gelu_exact(q.y + pv.y);
    }

#pragma unroll
    for (int nt = 0; nt < 4; ++nt) {
        const int n0 = nt * 16;
        v8f acc = zero8();
#pragma unroll
        for (int t = 0; t < 16; ++t) {
            const int k0 = 4 * t + 2 * half;
            v2f bw;
            bw.x = W2[(size_t)k0 * DD + n0 + lm];
            bw.y = W2[(size_t)(k0 + 1) * DD + n0 + lm];
            acc = __builtin_amdgcn_wmma_f32_16x16x4_f32(false, afrag[t], false, bw,
                                                        (short)0, acc, false, false);
        }
        const float bias = b2[n0 + lm];
        float m = -3.4e38f;
#pragma unroll
        for (int r = 0; r < 8; ++r) m = fmaxf(m, gelu_exact(acc[r] + bias));
        const float om = __shfl_xor(m, 16, 32);        // combine M=0..7 with M=8..15
        m = fmaxf(m, om);
        if (half == 0) out[(size_t)p * DD + n0 + lm] = m;
    }
}

// ---------------------------------------------------------------------------
extern "C" void kernel_launch(void* const* d_in, const int* in_sizes, int n_in,
                              void* d_out, int out_size, void* d_ws, size_t ws_size,
                              hipStream_t stream) {
    (void)in_sizes; (void)n_in; (void)out_size; (void)ws_size;
    const float* F  = (const float*)d_in[0];
    const float* W1 = (const float*)d_in[1];
    const float* b1 = (const float*)d_in[2];
    const float* W2 = (const float*)d_in[3];
    const float* b2 = (const float*)d_in[4];
    float* out = (float*)d_out;

    char* ws = (char*)d_ws;
    float* sq   = (float*)ws;                                   //  128 KB
    int*   idxb = (int*)(ws + 131072);                          //    2 MB
    float* P    = (float*)(ws + 131072 + 2097152);              //    8 MB
    float* Q    = (float*)(ws + 131072 + 2097152 + 8388608);    //    8 MB

    sqnorm_kernel  <<<BB * NN / 256, 256, 0, stream>>>(F, sq);
    knn_kernel     <<<BB * (NN / 16), 512, 0, stream>>>(F, sq, idxb);
    pq_kernel      <<<BB * (NN / 16), 128, 0, stream>>>(F, W1, b1, P, Q);
    edge_mlp_kernel<<<BB * NN / 8,   256, 0, stream>>>(P, Q, idxb, W2, b2, out);
}